// InvariantPointAttention_1915555414554
// MI455X (gfx1250) — compile-verified
//
#include <hip/hip_runtime.h>
#include <hip/hip_bf16.h>
#include <math.h>

// ---------------- problem constants ----------------
#define NRES   512
#define HEADS  12
#define DSS    16          // scalar head dim
#define DPQn   4           // query/key point dim
#define DPVn   8           // value point dim
#define DPAIR  128
#define DMODEL 384
#define DCAT   2112        // H*(DS + DP + DPV*4) = 12*176

#define SCALAR_SCALE 0.14433756729740643f   // (3*16)^-0.5
#define POINT_SCALE  0.13608276348795434f   // (3*4*4.5)^-0.5
#define PAIR_SCALE   0.5773502691896258f    // 3^-0.5
#define NEG_MAX      (-3.402823466e38f)
#define IPA_EPS      1e-8f

typedef float v2f __attribute__((ext_vector_type(2)));
typedef float v8f __attribute__((ext_vector_type(8)));

__device__ __forceinline__ v8f wmma_f32_k4(v2f a, v2f b, v8f c) {
  // D = A(16x4 f32) * B(4x16 f32) + C(16x16 f32)   [V_WMMA_F32_16X16X4_F32]
  return __builtin_amdgcn_wmma_f32_16x16x4_f32(
      /*neg_a=*/false, a, /*neg_b=*/false, b,
      /*c_mod=*/(short)0, c, /*reuse_a=*/false, /*reuse_b=*/false);
}

// ---------------- Tensor Data Mover (gfx1250) ----------------
#if __has_builtin(__builtin_amdgcn_tensor_load_to_lds) && \
    __has_builtin(__builtin_amdgcn_s_wait_tensorcnt)
#define HAVE_TDM 1
#else
#define HAVE_TDM 0
#endif

#if HAVE_TDM
typedef unsigned int u32x4 __attribute__((ext_vector_type(4)));
typedef int          i32x4 __attribute__((ext_vector_type(4)));
typedef int          i32x8 __attribute__((ext_vector_type(8)));

// Issue one TDM 2D tile load: tile_rows x DPAIR f32 tile from a row-major
// (tensor_rows x DPAIR) f32 tensor into LDS.  D# per CDNA5 ISA.
__device__ __forceinline__ void tdm_load_2d(unsigned int lds_addr,
                                            const float* gptr,
                                            unsigned int tensor_rows,
                                            unsigned int tile_rows) {
  const unsigned long long ga = (unsigned long long)(uintptr_t)gptr;
  const u32x4 g0 = {
      1u,                                             // count=1 (valid user D#)
      lds_addr,                                       // LDS byte address
      (unsigned int)ga,                               // global_addr[31:0]
      (unsigned int)((ga >> 32) & 0x01FFFFFFull) |    // global_addr[56:32]
          (2u << 30)                                  // type=2 ("image")
  };
  const i32x8 g1 = {
      (int)(2u << 16),                            // data_size=2 (4 bytes/elem)
      (int)(((unsigned)DPAIR & 0xffffu) << 16),   // tensor_dim0 = 128 (lo16)
      (int)((tensor_rows & 0xffffu) << 16),       // dim0 hi=0 | dim1 lo16
      (int)((tensor_rows >> 16) |                 // dim1 hi16
            ((unsigned)DPAIR << 16)),             // | tile_dim0 = 128
      (int)tile_rows,                             // tile_dim1, tile_dim2=0
      (int)DPAIR,                                 // tensor_dim0_stride lo32
      0, 0                                        // stride hi / dim1 stride
  };
  const i32x4 z4 = {0, 0, 0, 0};                  // groups 2/3 unused (2D)
#if defined(__clang_major__) && (__clang_major__ >= 23)
  const i32x8 z8 = {0, 0, 0, 0, 0, 0, 0, 0};
  __builtin_amdgcn_tensor_load_to_lds(g0, g1, z4, z4, z8, 0);
#else
  __builtin_amdgcn_tensor_load_to_lds(g0, g1, z4, z4, 0);
#endif
}
#endif

// ---------------------------------------------------------------------------
// Generic batched fp32 WMMA GEMM:  C[z] = A[z] (MxK) * B[z] (KxN) (+ bias)
// One wave per 16x16 C tile.  M multiple of 16; N guarded at the STORE only:
// out-of-range columns are clamped at the load (column j of C depends only on
// column j of B) — keeps EXEC all-1s through the WMMA chain per the ISA.
// K must be a multiple of 4 (true for every use here).
// ---------------------------------------------------------------------------
__global__ void gemm_wmma_f32(const float* __restrict__ A, int lda, long sA,
                              const float* __restrict__ B, int ldb, long sB,
                              float* __restrict__ C, int ldc, long sC,
                              int M, int Ncols, int K,
                              const float* __restrict__ bias) {
  const int z = blockIdx.z;
  A += (size_t)z * (size_t)sA;
  B += (size_t)z * (size_t)sB;
  C += (size_t)z * (size_t)sC;
  const int lane = threadIdx.x;
  const int half = lane >> 4;      // 0: K pair (k,k+1), 1: (k+2,k+3)
  const int l16  = lane & 15;
  const int row  = blockIdx.y * 16 + l16;            // A row (< M always)
  const int col  = blockIdx.x * 16 + l16;            // B col
  const bool colok = col < Ncols;
  const int colc = colok ? col : (Ncols - 1);        // clamped, discarded later

  const float* ap = A + (size_t)row * lda + half * 2;
  const float* bp = B + (size_t)(half * 2) * ldb + colc;
  const size_t bstep = (size_t)4 * ldb;

  v8f acc = {};
  for (int k = 0; k < K; k += 4) {
    v2f a; a.x = ap[0]; a.y = ap[1];                 // contiguous -> b64
    v2f b; b.x = bp[0]; b.y = bp[ldb];
    if (k + 32 < K) {                                // global_prefetch_b8
      __builtin_prefetch(ap + 32, 0, 3);
      __builtin_prefetch(bp + 8 * bstep, 0, 3);
    }
    acc = wmma_f32_k4(a, b, acc);
    ap += 4;
    bp += bstep;
  }
  if (!colok) return;
  const float bv = bias ? bias[col] : 0.0f;
#pragma unroll
  for (int r = 0; r < 8; ++r) {
    const int rr = blockIdx.y * 16 + r + half * 8;
    C[(size_t)rr * ldc + col] = acc[r] + bv;
  }
}

// ---------------------------------------------------------------------------
// Pair-bias pass: pb[(i,j),h] = pairwise[(i,j),:] @ W_pair[:,h]
// Flattened GEMM M=262144, K=128, N=12.  The 134 MB pairwise stream comes in
// as TDM 32x128-f32 slabs, double-buffered in LDS (2 x 16 KB); W_pair (6 KB)
// is parked in LDS once per block.  2 waves/block, 4 slabs/block pipeline.
// ---------------------------------------------------------------------------
#define PB_TROWS 32             // flattened rows per slab
#define PB_SL    4              // slabs per block
__global__ void __launch_bounds__(64)
pairbias_kernel(const float* __restrict__ pairwise,
                const float* __restrict__ W_pr,
                float* __restrict__ pb) {
  __shared__ float slab[2][PB_TROWS * DPAIR];   // 2 x 16 KB
  __shared__ float wsh[DPAIR * HEADS];          // 6 KB
  const int tid  = threadIdx.x;
  const int wave = tid >> 5;                    // 0..1 -> 16-row tile
  const int lane = tid & 31;
  const int half = lane >> 4, l16 = lane & 15;
  const int s0   = blockIdx.x * PB_SL;
  const bool hok = l16 < HEADS;
  const int hcol = hok ? l16 : (HEADS - 1);     // clamped head column

  for (int t = tid; t < DPAIR * HEADS; t += 64) wsh[t] = W_pr[t];

#if HAVE_TDM
  if (wave == 0)
    tdm_load_2d((unsigned int)(uintptr_t)&slab[0][0],
                pairwise + (size_t)s0 * PB_TROWS * DPAIR,
                NRES * NRES, PB_TROWS);
#endif
  __syncthreads();                              // wsh visible

  for (int t = 0; t < PB_SL; ++t) {
#if HAVE_TDM
    if (wave == 0) {
      if (t + 1 < PB_SL) {
        tdm_load_2d((unsigned int)(uintptr_t)&slab[(t + 1) & 1][0],
                    pairwise + (size_t)(s0 + t + 1) * PB_TROWS * DPAIR,
                    NRES * NRES, PB_TROWS);
        __builtin_amdgcn_s_wait_tensorcnt(1);   // slab t complete
      } else {
        __builtin_amdgcn_s_wait_tensorcnt(0);
      }
    }
    __syncthreads();
    const float* lp = &slab[t & 1][0];
#else
    {   // fallback: cooperative coalesced b128 staging, single buffer
      const float4* src =
          (const float4*)(pairwise + (size_t)(s0 + t) * PB_TROWS * DPAIR);
      float4* dst = (float4*)&slab[0][0];
      for (int q = tid; q < PB_TROWS * DPAIR / 4; q += 64) dst[q] = src[q];
    }
    __syncthreads();
    const float* lp = &slab[0][0];
#endif
    const float* arow = lp + (size_t)(wave * 16 + l16) * DPAIR + half * 2;
    const float* bcol = wsh + (size_t)(half * 2) * HEADS + hcol;
    v8f acc = {};
#pragma unroll
    for (int kk = 0; kk < DPAIR; kk += 4) {
      v2f a; a.x = arow[kk]; a.y = arow[kk + 1];             // ds_load_b64
      v2f b; b.x = bcol[kk * HEADS]; b.y = bcol[kk * HEADS + HEADS];
      acc = wmma_f32_k4(a, b, acc);
    }
    const size_t grow0 = (size_t)(s0 + t) * PB_TROWS + wave * 16;
    if (hok) {
#pragma unroll
      for (int r = 0; r < 8; ++r)
        pb[(grow0 + r + half * 8) * HEADS + l16] = acc[r];
    }
    __syncthreads();                            // done reading this buffer
  }
}

// ---------------------------------------------------------------------------
// Apply per-residue frame to raw point projections:
//   out[n,(h,d),r] = sum_c raw[n,(h,d),c] * R[n,c,r] + t[n,r]
// ---------------------------------------------------------------------------
__global__ void point_frame_fwd(const float* __restrict__ raw,
                                const float* __restrict__ R,
                                const float* __restrict__ t,
                                float* __restrict__ out, int dnum) {
  const int idx = blockIdx.x * blockDim.x + threadIdx.x;
  const int total = NRES * HEADS * dnum;
  if (idx >= total) return;
  const int n  = idx / (HEADS * dnum);
  const int hd = idx % (HEADS * dnum);
  const float* p  = raw + (size_t)n * (HEADS * dnum * 3) + hd * 3;
  const float* Rn = R + (size_t)n * 9;
  const float* tn = t + (size_t)n * 3;
  const float c0 = p[0], c1 = p[1], c2 = p[2];
  float* o = out + (size_t)n * (HEADS * dnum * 3) + hd * 3;
#pragma unroll
  for (int r = 0; r < 3; ++r)
    o[r] = c0 * Rn[0 * 3 + r] + c1 * Rn[1 * 3 + r] + c2 * Rn[2 * 3 + r] + tn[r];
}

// q2[h,n] = sum_{d,c} qp_g[n,(h,d,c)]^2   (same for k2)
__global__ void sqnorm_kernel(const float* __restrict__ qpg,
                              const float* __restrict__ kpg,
                              float* __restrict__ q2, float* __restrict__ k2) {
  const int idx = blockIdx.x * blockDim.x + threadIdx.x;
  if (idx >= HEADS * NRES) return;
  const int h = idx / NRES, n = idx % NRES;
  const float* q  = qpg + (size_t)n * (HEADS * DPQn * 3) + h * (DPQn * 3);
  const float* kk = kpg + (size_t)n * (HEADS * DPQn * 3) + h * (DPQn * 3);
  float sq = 0.f, sk = 0.f;
#pragma unroll
  for (int e = 0; e < DPQn * 3; ++e) { sq += q[e] * q[e]; sk += kk[e] * kk[e]; }
  q2[h * NRES + n] = sq;
  k2[h * NRES + n] = sk;
}

// ---------------------------------------------------------------------------
// logits[h,i,j] = SCALAR_SCALE * <qs_i,ks_j>  +  PAIR_SCALE*(pair_bias+b_pair)
//              + pwc * <qp_i,kp_j> - 0.5*pwc*(q2_i + k2_j),  masked.
// One wave per 16x16 (i,j) tile per head; two WMMA chains (K=16 and K=12).
// ---------------------------------------------------------------------------
__global__ void logits_kernel(const float* __restrict__ qs,
                              const float* __restrict__ ks,
                              const float* __restrict__ qpg,
                              const float* __restrict__ kpg,
                              const float* __restrict__ q2,
                              const float* __restrict__ k2,
                              const float* __restrict__ pairbias,
                              const float* __restrict__ b_pair,
                              const float* __restrict__ point_weights,
                              const unsigned char* __restrict__ mask,
                              float* __restrict__ logits) {
  const int h = blockIdx.z;
  const int i0 = blockIdx.y * 16, j0 = blockIdx.x * 16;
  const int lane = threadIdx.x;
  const int half = lane >> 4, l16 = lane & 15;
  const int row = i0 + l16, col = j0 + l16;

  const float* qsp = qs + (size_t)row * (HEADS * DSS) + h * DSS + half * 2;
  const float* ksp = ks + (size_t)col * (HEADS * DSS) + h * DSS + half * 2;
  v8f accS = {};
#pragma unroll
  for (int k = 0; k < DSS; k += 4) {
    v2f a; a.x = qsp[k]; a.y = qsp[k + 1];
    v2f b; b.x = ksp[k]; b.y = ksp[k + 1];
    accS = wmma_f32_k4(a, b, accS);
  }
  const float* qpp = qpg + (size_t)row * (HEADS * DPQn * 3) + h * (DPQn * 3) + half * 2;
  const float* kpp = kpg + (size_t)col * (HEADS * DPQn * 3) + h * (DPQn * 3) + half * 2;
  v8f accP = {};
#pragma unroll
  for (int k = 0; k < DPQn * 3; k += 4) {
    v2f a; a.x = qpp[k]; a.y = qpp[k + 1];
    v2f b; b.x = kpp[k]; b.y = kpp[k + 1];
    accP = wmma_f32_k4(a, b, accP);
  }
  const float pw = point_weights[h];
  const float sp = (pw > 20.f) ? pw : log1pf(__expf(pw));  // softplus
  const float pwc = sp * POINT_SCALE;
  const float bph = b_pair[h];
  const float k2j = k2[h * NRES + col];
#pragma unroll
  for (int r = 0; r < 8; ++r) {
    const int i = i0 + r + half * 8;
    const int j = col;
    float v = SCALAR_SCALE * accS[r] + pwc * accP[r]
            - 0.5f * pwc * (q2[h * NRES + i] + k2j)
            + PAIR_SCALE * (pairbias[((size_t)i * NRES + j) * HEADS + h] + bph);
    if (!(mask[i] && mask[j])) v = NEG_MAX;
    logits[((size_t)h * NRES + i) * NRES + j] = v;
  }
}

// Row softmax over 512 elems: one wave per (h,i) row, 16 elems/lane.
__global__ void softmax_kernel(float* __restrict__ attn) {
  float* p = attn + (size_t)blockIdx.x * NRES;
  const int lane = threadIdx.x;
  float vals[16];
  float m = NEG_MAX;
#pragma unroll
  for (int t = 0; t < 16; ++t) { vals[t] = p[lane + t * 32]; m = fmaxf(m, vals[t]); }
#pragma unroll
  for (int off = 16; off > 0; off >>= 1) m = fmaxf(m, __shfl_xor(m, off, 32));
  float s = 0.f;
#pragma unroll
  for (int t = 0; t < 16; ++t) { vals[t] = __expf(vals[t] - m); s += vals[t]; }
#pragma unroll
  for (int off = 16; off > 0; off >>= 1) s += __shfl_xor(s, off, 32);
  const float inv = 1.f / s;
#pragma unroll
  for (int t = 0; t < 16; ++t) p[lane + t * 32] = vals[t] * inv;
}

// ---------------------------------------------------------------------------
// res_pair[h,i,:] = attn[h,i,:] @ pairwise[i,:,:]  (K=512, per-residue GEMM).
// One workgroup per residue i; the 256 KB pairwise[i] slab is streamed once
// via TDM into a double-buffered LDS panel (2 x 16 KB), wave 0 driving DMA +
// s_wait_tensorcnt, all 8 waves feeding WMMA from LDS.  A rows = 12 heads,
// clamped to 16.  Output lands in concat buffer at offset 576 + h*128 + d.
// ---------------------------------------------------------------------------
#define KPAN 32
#define NPAN (NRES / KPAN)
__global__ void __launch_bounds__(256)
respair_kernel(const float* __restrict__ attn,
               const float* __restrict__ pairwise,
               float* __restrict__ o) {
  __shared__ float panel[2][KPAN * DPAIR];
  const int i    = blockIdx.x;
  const int tid  = threadIdx.x;
  const int wave = tid >> 5;               // 0..7 -> 16-channel block
  const int lane = tid & 31;
  const int half = lane >> 4, l16 = lane & 15;
  const int ncol = wave * 16 + l16;        // 0..127 pair channel
  const int m    = (l16 < HEADS) ? l16 : (HEADS - 1);   // clamped head row
  const float* Ap   = attn + ((size_t)m * NRES + i) * NRES + half * 2;
  const float* gsrc = pairwise + (size_t)i * NRES * DPAIR;

#if HAVE_TDM
  if (wave == 0)
    tdm_load_2d((unsigned int)(uintptr_t)&panel[0][0], gsrc, NRES, KPAN);
#endif

  v8f acc = {};
  for (int p = 0; p < NPAN; ++p) {
#if HAVE_TDM
    if (wave == 0) {
      if (p + 1 < NPAN) {
        tdm_load_2d((unsigned int)(uintptr_t)&panel[(p + 1) & 1][0],
                    gsrc + (size_t)(p + 1) * KPAN * DPAIR, NRES, KPAN);
        __builtin_amdgcn_s_wait_tensorcnt(1);   // panel p complete
      } else {
        __builtin_amdgcn_s_wait_tensorcnt(0);
      }
    }
    __syncthreads();                            // panel p visible to all waves
    const float* lp = &panel[p & 1][0];
#else
    {   // fallback: cooperative coalesced b128 staging, single buffer
      const float4* src = (const float4*)(gsrc + (size_t)p * KPAN * DPAIR);
      float4* dst = (float4*)&panel[0][0];
      for (int t = tid; t < KPAN * DPAIR / 4; t += 256) dst[t] = src[t];
    }
    __syncthreads();
    const float* lp = &panel[0][0];
#endif
    const float* apk = Ap + p * KPAN;
#pragma unroll
    for (int kk = 0; kk < KPAN; kk += 4) {
      const int ka = kk + half * 2;
      v2f a; a.x = apk[kk]; a.y = apk[kk + 1];
      v2f b; b.x = lp[ka * DPAIR + ncol]; b.y = lp[(ka + 1) * DPAIR + ncol];
      acc = wmma_f32_k4(a, b, acc);
    }
    __syncthreads();                            // done reading this buffer
  }
#pragma unroll
  for (int r = 0; r < 8; ++r) {
    const int hh = r + half * 8;
    if (hh < HEADS)
      o[(size_t)i * DCAT + 576 + hh * DPAIR + ncol] = acc[r];
  }
}

// ---------------------------------------------------------------------------
// Invert frame on aggregated points, write local coords + norms into concat:
//   loc[r] = sum_c (g[c]-t[i,c]) * R[i,r,c]
// ---------------------------------------------------------------------------
__global__ void finalize_points(const float* __restrict__ pts_g,
                                const float* __restrict__ R,
                                const float* __restrict__ t,
                                float* __restrict__ o) {
  const int idx = blockIdx.x * blockDim.x + threadIdx.x;
  const int total = NRES * HEADS * DPVn;
  if (idx >= total) return;
  const int n  = idx / (HEADS * DPVn);
  const int hd = idx % (HEADS * DPVn);
  const int h = hd / DPVn, d = hd % DPVn;
  const float* g  = pts_g + (size_t)n * (HEADS * DPVn * 3) + hd * 3;
  const float* Rn = R + (size_t)n * 9;
  const float* tn = t + (size_t)n * 3;
  const float d0 = g[0] - tn[0], d1 = g[1] - tn[1], d2 = g[2] - tn[2];
  float s = 0.f;
  float* op = o + (size_t)n * DCAT + 192 + h * (DPVn * 3) + d * 3;
#pragma unroll
  for (int r = 0; r < 3; ++r) {
    const float lr = d0 * Rn[r * 3 + 0] + d1 * Rn[r * 3 + 1] + d2 * Rn[r * 3 + 2];
    op[r] = lr;
    s += lr * lr;
  }
  o[(size_t)n * DCAT + 480 + h * DPVn + d] = sqrtf(s + IPA_EPS);
}

// ---------------------------------------------------------------------------
extern "C" void kernel_launch(void* const* d_in, const int* in_sizes, int n_in,
                              void* d_out, int out_size, void* d_ws, size_t ws_size,
                              hipStream_t stream) {
  const float* x    = (const float*)d_in[0];   // (512,384)
  const float* pair = (const float*)d_in[1];   // (512,512,128)
  const float* rot  = (const float*)d_in[2];   // (512,3,3)
  const float* trn  = (const float*)d_in[3];   // (512,3)
  const unsigned char* mask = (const unsigned char*)d_in[4];  // (512) bool
  const float* W_qs = (const float*)d_in[5];   // (384,192)
  const float* W_ks = (const float*)d_in[6];
  const float* W_vs = (const float*)d_in[7];
  const float* W_qp = (const float*)d_in[8];   // (384,144)
  const float* W_kp = (const float*)d_in[9];
  const float* W_vp = (const float*)d_in[10];  // (384,288)
  const float* W_pr = (const float*)d_in[11];  // (128,12)
  const float* b_pr = (const float*)d_in[12];  // (12)
  const float* ptw  = (const float*)d_in[13];  // (12)
  const float* W_o  = (const float*)d_in[14];  // (2112,384)
  const float* b_o  = (const float*)d_in[15];  // (384)
  float* out = (float*)d_out;                  // (512,384)
  float* ws  = (float*)d_ws;

  // workspace layout (floats)
  float* qs_b   = ws;                         // 512*192
  float* ks_b   = qs_b   + 512 * 192;
  float* vs_b   = ks_b   + 512 * 192;
  float* qp_raw = vs_b   + 512 * 192;         // 512*144
  float* kp_raw = qp_raw + 512 * 144;
  float* vp_raw = kp_raw + 512 * 144;         // 512*288
  float* qp_g   = vp_raw + 512 * 288;         // 512*144
  float* kp_g   = qp_g   + 512 * 144;
  float* vp_g   = kp_g   + 512 * 144;         // 512*288
  float* q2_b   = vp_g   + 512 * 288;         // 12*512
  float* k2_b   = q2_b   + HEADS * 512;
  float* pb_b   = k2_b   + HEADS * 512;       // 512*512*12
  float* attn_b = pb_b   + (size_t)512 * 512 * HEADS;  // 12*512*512
  float* pts_g  = attn_b + (size_t)HEADS * 512 * 512;  // 512*288
  float* cat_b  = pts_g  + 512 * 288;         // 512*2112

  const dim3 w(32);

  // 1) scalar + point projections (fp32 WMMA GEMMs)
  gemm_wmma_f32<<<dim3(12, 32, 1), w, 0, stream>>>(x, DMODEL, 0, W_qs, 192, 0,
      qs_b, 192, 0, NRES, 192, DMODEL, nullptr);
  gemm_wmma_f32<<<dim3(12, 32, 1), w, 0, stream>>>(x, DMODEL, 0, W_ks, 192, 0,
      ks_b, 192, 0, NRES, 192, DMODEL, nullptr);
  gemm_wmma_f32<<<dim3(12, 32, 1), w, 0, stream>>>(x, DMODEL, 0, W_vs, 192, 0,
      vs_b, 192, 0, NRES, 192, DMODEL, nullptr);
  gemm_wmma_f32<<<dim3(9, 32, 1),  w, 0, stream>>>(x, DMODEL, 0, W_qp, 144, 0,
      qp_raw, 144, 0, NRES, 144, DMODEL, nullptr);
  gemm_wmma_f32<<<dim3(9, 32, 1),  w, 0, stream>>>(x, DMODEL, 0, W_kp, 144, 0,
      kp_raw, 144, 0, NRES, 144, DMODEL, nullptr);
  gemm_wmma_f32<<<dim3(18, 32, 1), w, 0, stream>>>(x, DMODEL, 0, W_vp, 288, 0,
      vp_raw, 288, 0, NRES, 288, DMODEL, nullptr);

  // 2) rotate + translate points into the global frame
  point_frame_fwd<<<(NRES * HEADS * DPQn + 255) / 256, 256, 0, stream>>>(
      qp_raw, rot, trn, qp_g, DPQn);
  point_frame_fwd<<<(NRES * HEADS * DPQn + 255) / 256, 256, 0, stream>>>(
      kp_raw, rot, trn, kp_g, DPQn);
  point_frame_fwd<<<(NRES * HEADS * DPVn + 255) / 256, 256, 0, stream>>>(
      vp_raw, rot, trn, vp_g, DPVn);
  sqnorm_kernel<<<(HEADS * NRES + 255) / 256, 256, 0, stream>>>(qp_g, kp_g,
      q2_b, k2_b);

  // 3) pair bias — first full pairwise pass, TDM-staged through LDS
  pairbias_kernel<<<dim3((NRES * NRES) / (PB_TROWS * PB_SL)), 64, 0, stream>>>(
      pair, W_pr, pb_b);

  // 4) fused logits (scalar WMMA + point WMMA + bias + dist terms + mask)
  logits_kernel<<<dim3(32, 32, HEADS), w, 0, stream>>>(qs_b, ks_b, qp_g, kp_g,
      q2_b, k2_b, pb_b, b_pr, ptw, mask, attn_b);

  // 5) softmax rows
  softmax_kernel<<<HEADS * NRES, w, 0, stream>>>(attn_b);

  // 6) attn @ vs -> concat[.,0:192]   (batched over heads)
  gemm_wmma_f32<<<dim3(1, 32, HEADS), w, 0, stream>>>(
      attn_b, NRES, (long)NRES * NRES, vs_b, 192, 16,
      cat_b, DCAT, 16, NRES, DSS, NRES, nullptr);

  // 7) attn @ vp_g -> global-frame aggregated points (batched over heads)
  gemm_wmma_f32<<<dim3(2, 32, HEADS), w, 0, stream>>>(
      attn_b, NRES, (long)NRES * NRES, vp_g, 288, 24,
      pts_g, 288, 24, NRES, DPVn * 3, NRES, nullptr);

  // 8) attn @ pairwise -> concat[.,576:2112] — second full pass, TDM-staged
  respair_kernel<<<NRES, 256, 0, stream>>>(attn_b, pair, cat_b);

  // 9) invert frames, write local points + norms into concat
  finalize_points<<<(NRES * HEADS * DPVn + 255) / 256, 256, 0, stream>>>(
      pts_g, rot, trn, cat_b);

  // 10) output projection: (512,2112) @ (2112,384) + b_out
  gemm_wmma_f32<<<dim3(24, 32, 1), w, 0, stream>>>(
      cat_b, DCAT, 0, W_o, DMODEL, 0, out, DMODEL, 0,
      NRES, DMODEL, DCAT, b_o);
}